// SubMConv3dTestTorch_26963804684443
// MI455X (gfx1250) — compile-verified
//
#include <hip/hip_runtime.h>

// ---------------------------------------------------------------------------
// Submanifold sparse conv x2 for MI455X (gfx1250), fp32 WMMA path with
// async (ASYNCcnt-tracked) double-buffered weight staging into LDS.
// ---------------------------------------------------------------------------
#define BB   2
#define DD   15
#define HH   400
#define WWID 400
#define TBL  (BB * DD * HH * WWID)   // 4,800,000 voxel hash entries
#define KVOL 27
#define CO   64

typedef __attribute__((ext_vector_type(2))) float v2f;
typedef __attribute__((ext_vector_type(8))) float v8f;
typedef __attribute__((__vector_size__(16))) int vi4;
typedef __attribute__((address_space(1))) vi4 g_vi4;   // global int4
typedef __attribute__((address_space(3))) vi4 l_vi4;   // LDS int4

#if __has_builtin(__builtin_amdgcn_global_load_async_to_lds_b128)
#define USE_ASYNC_LDS 1
#else
#define USE_ASYNC_LDS 0
#endif

// ---------------- rulebook construction ----------------

__global__ __launch_bounds__(256) void fill_neg1(int* __restrict__ t, int n) {
  int i = blockIdx.x * 256 + threadIdx.x;
  if (i < n) t[i] = -1;
}

__global__ __launch_bounds__(256) void scatter_idx(const int* __restrict__ coors,
                                                   int* __restrict__ tbl, int n) {
  int i = blockIdx.x * 256 + threadIdx.x;
  if (i >= n) return;
  int b = coors[4 * i + 0];
  int z = coors[4 * i + 1];
  int y = coors[4 * i + 2];
  int x = coors[4 * i + 3];
  long flat = (((long)b * DD + z) * HH + y) * WWID + x;
  tbl[flat] = i;
}

__global__ __launch_bounds__(256) void build_nbr(const int* __restrict__ coors,
                                                 const int* __restrict__ tbl,
                                                 int* __restrict__ nbr, int n) {
  int i = blockIdx.x * 256 + threadIdx.x;
  if (i >= n) return;
  int b = coors[4 * i + 0];
  int z = coors[4 * i + 1];
  int y = coors[4 * i + 2];
  int x = coors[4 * i + 3];
#pragma unroll
  for (int k = 0; k < KVOL; ++k) {
    const int dz = k / 9 - 1;
    const int dy = (k / 3) % 3 - 1;
    const int dx = k % 3 - 1;
    int nz = z + dz, ny = y + dy, nx = x + dx;
    int v = -1;
    if (nz >= 0 && nz < DD && ny >= 0 && ny < HH && nx >= 0 && nx < WWID) {
      long flat = (((long)b * DD + nz) * HH + ny) * WWID + nx;
      v = tbl[flat];
    }
    nbr[(long)k * n + i] = v;
  }
}

// ---------------- async helpers ----------------

template <int N>
__device__ __forceinline__ void wait_asynccnt() {
#if USE_ASYNC_LDS
#if __has_builtin(__builtin_amdgcn_s_wait_asynccnt)
  __builtin_amdgcn_s_wait_asynccnt(N);
#else
  asm volatile("s_wait_asynccnt %0" ::"i"(N) : "memory");
#endif
#endif
}

// Stage NBYTES of weights global->LDS with NT threads; async path copies 16B
// per lane per issue (NBYTES/(NT*16) async instructions per wave).
template <int NBYTES, int NT>
__device__ __forceinline__ void stage_weights(const float* __restrict__ src,
                                              float* __restrict__ dst, int tid) {
  constexpr int NV4 = NBYTES / 16;
#if USE_ASYNC_LDS
  g_vi4* g = (g_vi4*)src;   // builtin takes non-const global int4*
  l_vi4* l = (l_vi4*)dst;   // LDS int4*
#pragma unroll
  for (int it = 0; it < NV4 / NT; ++it) {
    const int idx = tid + it * NT;
    __builtin_amdgcn_global_load_async_to_lds_b128(g + idx, l + idx, 0, 0);
  }
#else
  const float4* s4 = (const float4*)src;
  float4* d4 = (float4*)dst;
#pragma unroll
  for (int it = 0; it < NV4 / NT; ++it) {
    const int idx = tid + it * NT;
    d4[idx] = s4[idx];
  }
#endif
}

// ---------------- gather-GEMM layer: out[i] = sum_k W[k]^T feats[nbr[k,i]] ----
//
// Block = 256 threads = 8 waves. Each wave owns a 16-voxel row tile and all 64
// output channels as four 16x16 f32 accumulator tiles. W[k] (CIN x 64) is
// double-buffered in LDS: the k+1 tile streams in asynchronously while the
// 4*CIN/4 V_WMMA_F32_16X16X4_F32 ops of iteration k execute.

template <int CIN>
__global__ __launch_bounds__(256)
void subm_conv_wmma(const float* __restrict__ feats,   // [n, CIN]
                    const float* __restrict__ wall,    // [27, CIN, 64]
                    const int* __restrict__ nbr,       // [27, n]
                    float* __restrict__ out,           // [n, 64]
                    int n) {
  constexpr int NCH   = CIN / 4;              // K chunks of 4
  constexpr int WB    = CIN * CO;             // floats per weight tile
  constexpr int NSTG  = (WB * 4) / (256 * 16);// async issues per wave per stage
  __shared__ float lw[2][WB];

  const int tid  = threadIdx.x;
  const int lane = tid & 31;
  const int wv   = tid >> 5;                  // wave in block, 0..7
  const int m    = lane & 15;                 // row / column within 16-tile
  const int hs   = lane >> 4;                 // lane half-select
  const int p    = hs * 2;                    // K sub-offset for A/B fragments
  const long base = (long)blockIdx.x * 128 + (long)wv * 16;

  v8f acc0 = {}; v8f acc1 = {}; v8f acc2 = {}; v8f acc3 = {};

  // prologue: stage k=0 into buffer 0
  stage_weights<WB * 4, 256>(wall, lw[0], tid);

  for (int k = 0; k < KVOL; ++k) {
    const int cur = k & 1;
    if (k + 1 < KVOL) {
      stage_weights<WB * 4, 256>(wall + (long)(k + 1) * WB, lw[cur ^ 1], tid);
      wait_asynccnt<NSTG>();                  // k's tile has landed (per wave)
    } else {
      wait_asynccnt<0>();
    }
    __syncthreads();                          // all waves see lw[cur]

    // ---- gather this wave's 16 feature rows into A-fragment layout ----
    const long row = base + m;
    const int  j   = (row < n) ? nbr[(long)k * n + row] : -1;
    const float* fr = feats + (long)(j < 0 ? 0 : j) * CIN;
    const v2f zz = {0.0f, 0.0f};
    v2f a[NCH];
#pragma unroll
    for (int c = 0; c < NCH; ++c) {
      v2f t = *(const v2f*)(fr + 4 * c + p);
      a[c] = (j < 0) ? zz : t;                // per-lane select, EXEC stays full
    }

    // ---- K-chunk loop: 4 output tiles x V_WMMA_F32_16X16X4_F32 ----
    const float* wb = lw[cur];
#pragma unroll
    for (int c = 0; c < NCH; ++c) {
      const int kr = 4 * c + p;               // B rows kr, kr+1 for this lane half
      v2f b;
      b.x = wb[kr * CO + 0 + m];
      b.y = wb[(kr + 1) * CO + 0 + m];
      acc0 = __builtin_amdgcn_wmma_f32_16x16x4_f32(false, a[c], false, b,
                                                   (short)0, acc0, false, false);
      b.x = wb[kr * CO + 16 + m];
      b.y = wb[(kr + 1) * CO + 16 + m];
      acc1 = __builtin_amdgcn_wmma_f32_16x16x4_f32(false, a[c], false, b,
                                                   (short)0, acc1, false, false);
      b.x = wb[kr * CO + 32 + m];
      b.y = wb[(kr + 1) * CO + 32 + m];
      acc2 = __builtin_amdgcn_wmma_f32_16x16x4_f32(false, a[c], false, b,
                                                   (short)0, acc2, false, false);
      b.x = wb[kr * CO + 48 + m];
      b.y = wb[(kr + 1) * CO + 48 + m];
      acc3 = __builtin_amdgcn_wmma_f32_16x16x4_f32(false, a[c], false, b,
                                                   (short)0, acc3, false, false);
    }

    __syncthreads();   // everyone done reading lw[cur]; safe to overwrite next
  }

  // ---- store: C/D layout -> VGPR r holds row r + 8*hs, column lane&15 ----
#pragma unroll
  for (int r = 0; r < 8; ++r) {
    const long row = base + r + 8 * hs;
    if (row < n) {
      float* o = out + row * CO + m;
      o[0]  = acc0[r];
      o[16] = acc1[r];
      o[32] = acc2[r];
      o[48] = acc3[r];
    }
  }
}

// ---------------- launch ----------------

extern "C" void kernel_launch(void* const* d_in, const int* in_sizes, int n_in,
                              void* d_out, int out_size, void* d_ws, size_t ws_size,
                              hipStream_t stream) {
  (void)n_in; (void)out_size; (void)ws_size;
  const float* feats = (const float*)d_in[0];   // [n, 32] f32
  const int*   coors = (const int*)d_in[1];     // [n, 4] i32
  const float* W1    = (const float*)d_in[3];   // [27, 32, 64] f32
  const float* W2    = (const float*)d_in[4];   // [27, 64, 64] f32
  float* out = (float*)d_out;                   // [n, 64] f32

  const int n = in_sizes[0] / 32;

  // workspace layout: [hash table][rulebook][layer-1 output]
  char* ws = (char*)d_ws;
  int* tbl = (int*)ws;
  size_t nbr_off = (size_t)TBL * 4;
  int* nbr = (int*)(ws + nbr_off);
  size_t mid_off = (nbr_off + (size_t)KVOL * (size_t)n * 4 + 15) & ~(size_t)15;
  float* mid = (float*)(ws + mid_off);

  fill_neg1  <<<(TBL + 255) / 256, 256, 0, stream>>>(tbl, TBL);
  scatter_idx<<<(n + 255) / 256,   256, 0, stream>>>(coors, tbl, n);
  build_nbr  <<<(n + 255) / 256,   256, 0, stream>>>(coors, tbl, nbr, n);

  const int nblk = (n + 127) / 128;
  subm_conv_wmma<32><<<nblk, 256, 0, stream>>>(feats, W1, nbr, mid, n);
  subm_conv_wmma<64><<<nblk, 256, 0, stream>>>(mid,   W2, nbr, out, n);
}